// GeodesicLayer_67121748902279
// MI455X (gfx1250) — compile-verified
//
#include <hip/hip_runtime.h>
#include <hip/hip_bf16.h>
#include <math.h>

// GeodesicLayer on MI455X (gfx1250, wave32).
// Phase 0: tiny prep kernel (p, p2, s, a_norm).
// Phase 1: X * P^T GEMM via V_WMMA_F32_16X16X4_F32 with LDS-staged B matrix
//          (pitch-68 layout -> conflict-free ds_load_b64) and fused epilogue.
// Memory-bound (~20 MB traffic); fp32 WMMA is exact and more than fast enough.

typedef float v2f __attribute__((ext_vector_type(2)));
typedef float v8f __attribute__((ext_vector_type(8)));

#define EPSF 1e-15f
#define LDS_PITCH 68   // 64 + 4 dwords: lane->bank stride 4, conflict-free b64 reads

// ---------------- Phase 0: per-output parameters ----------------
// p[o,i] = w[o,i]*(1-b_o^2); p2[o] = ||p[o]||^2; u = p*b;
// un = max(|b|*||p||, eps); s_o = tanh(un)*b/un; a = s_o * p
// a_norm = max(sqrt(sum_o s_o^2 * p2[o]), eps)
__global__ void __launch_bounds__(256)
geo_prep(const float* __restrict__ w_raw, const float* __restrict__ b_raw,
         float* __restrict__ p, float* __restrict__ p2s,
         float* __restrict__ ss, float* __restrict__ anorm) {
    __shared__ float red[256];
    const int o = threadIdx.x;           // 0..255
    const float b = b_raw[o];
    const float f = 1.0f - b * b;
    float sq = 0.0f;
#pragma unroll 8
    for (int i = 0; i < 64; ++i) {
        float v = w_raw[o * 64 + i] * f;
        p[o * 64 + i] = v;
        sq += v * v;
    }
    p2s[o] = sq;
    float un = fmaxf(sqrtf(sq) * fabsf(b), EPSF);
    float s = tanhf(un) * b / un;
    ss[o] = s;
    red[o] = s * s * sq;   // ||a[o]||^2
    __syncthreads();
#pragma unroll
    for (int off = 128; off > 0; off >>= 1) {
        if (o < off) red[o] += red[o + off];
        __syncthreads();
    }
    if (o == 0) anorm[0] = fmaxf(sqrtf(red[0]), EPSF);
}

// ---------------- Phase 1: GEMM + fused epilogue ----------------
// Block: 128 threads (4 waves). Whole B matrix (256x64 f32) staged in LDS once.
// Each wave: two 16-row strips of X (A frags in regs), loops over 16 N-tiles;
// per tile: 16 ds_load_b64 (clause) -> one wait -> 32 back-to-back WMMAs.
__global__ void __launch_bounds__(128)
geo_main(const float* __restrict__ X,      // [B,64]
         const float* __restrict__ P,      // [256,64]
         const float* __restrict__ p2s,    // [256]
         const float* __restrict__ ss,     // [256]
         const float* __restrict__ anormp, // [1]
         float* __restrict__ out) {        // [B,256]
    __shared__ float Plds[256 * LDS_PITCH];   // ~68 KB of the WGP's 320 KB

    const int t = threadIdx.x;
    // ---- cooperative stage of P into LDS (row-major, pitch 68 dwords) ----
#pragma unroll
    for (int i = 0; i < 32; ++i) {
        const int j   = i * 128 + t;      // float4 id, 0..4095
        const int row = j >> 4;           // 16 float4 per 64-float row
        const int c4  = j & 15;
        const float4 v = ((const float4*)P)[j];
        *(float4*)&Plds[row * LDS_PITCH + c4 * 4] = v;
    }

    const int lane = t & 31;
    const int wid  = t >> 5;
    const int gwave = blockIdx.x * 4 + wid;
    const int lm = lane & 15;   // element row/col within half-wave
    const int lh = lane >> 4;   // half-wave id
    const float anorm = anormp[0];

    const int row0 = gwave * 32;              // two strips: rows row0.. and row0+16..

    // --- load A fragments for both strips; per-lane x2 partial sums ---
    v2f a0[16], a1[16];
    float xs0 = 0.0f, xs1 = 0.0f;
    const float* A0 = X + (size_t)(row0 + lm) * 64 + 2 * lh;
    const float* A1 = X + (size_t)(row0 + 16 + lm) * 64 + 2 * lh;
#pragma unroll
    for (int k = 0; k < 16; ++k) {
        v2f t0 = *(const v2f*)(A0 + 4 * k);
        v2f t1 = *(const v2f*)(A1 + 4 * k);
        a0[k] = t0; a1[k] = t1;
        xs0 += t0.x * t0.x + t0.y * t0.y;
        xs1 += t1.x * t1.x + t1.y * t1.y;
    }
    // lanes L and L^16 together cover all K=64 for row L%16
    xs0 += __shfl_xor(xs0, 16, 32);
    xs1 += __shfl_xor(xs1, 16, 32);
    // x2 for the row each accumulator VGPR v maps to: m = v + 8*lh
    float rx0[8], rx1[8];
#pragma unroll
    for (int v = 0; v < 8; ++v) {
        int src = v + 8 * lh;
        rx0[v] = __shfl(xs0, src, 32);
        rx1[v] = __shfl(xs1, src, 32);
    }

    __syncthreads();   // P staged

    for (int nt = 0; nt < 16; ++nt) {
        const int col = nt * 16 + lm;       // output column this lane touches
        const float* Bp = &Plds[col * LDS_PITCH + 2 * lh];  // B[k,n] = P[n,k]

        // ---- all 16 B fragments first (ds clause, single wait) ----
        v2f bf[16];
#pragma unroll
        for (int k = 0; k < 16; ++k) {
            bf[k] = *(const v2f*)(Bp + 4 * k);
        }

        v8f c0 = {0.f,0.f,0.f,0.f,0.f,0.f,0.f,0.f};
        v8f c1 = {0.f,0.f,0.f,0.f,0.f,0.f,0.f,0.f};
#pragma unroll
        for (int k = 0; k < 16; ++k) {
            c0 = __builtin_amdgcn_wmma_f32_16x16x4_f32(
                    false, a0[k], false, bf[k], (short)0, c0, false, false);
            c1 = __builtin_amdgcn_wmma_f32_16x16x4_f32(
                    false, a1[k], false, bf[k], (short)0, c1, false, false);
        }

        const float p2   = p2s[col];
        const float s    = ss[col];
        const float beta = 1.0f - p2;

#pragma unroll
        for (int v = 0; v < 8; ++v) {
            // strip 0
            {
                float G  = c0[v];
                float x2 = rx0[v];
                float alpha = 1.0f - 2.0f * G + x2;
                float den   = fmaxf(1.0f - 2.0f * G + p2 * x2, EPSF);
                float rden  = 1.0f / den;
                float mob2  = fmaxf((alpha * alpha * p2 - 2.0f * alpha * beta * G
                                     + beta * beta * x2) * rden * rden, EPSF);
                float sc    = s * (beta * G - alpha * p2) * rden;
                float denom = fmaxf((1.0f - mob2) * anorm, EPSF);
                float z     = 2.0f * sc / denom;
                int r = row0 + v + 8 * lh;
                out[(size_t)r * 256 + col] = asinhf(z) * anorm;
            }
            // strip 1
            {
                float G  = c1[v];
                float x2 = rx1[v];
                float alpha = 1.0f - 2.0f * G + x2;
                float den   = fmaxf(1.0f - 2.0f * G + p2 * x2, EPSF);
                float rden  = 1.0f / den;
                float mob2  = fmaxf((alpha * alpha * p2 - 2.0f * alpha * beta * G
                                     + beta * beta * x2) * rden * rden, EPSF);
                float sc    = s * (beta * G - alpha * p2) * rden;
                float denom = fmaxf((1.0f - mob2) * anorm, EPSF);
                float z     = 2.0f * sc / denom;
                int r = row0 + 16 + v + 8 * lh;
                out[(size_t)r * 256 + col] = asinhf(z) * anorm;
            }
        }
    }
}

extern "C" void kernel_launch(void* const* d_in, const int* in_sizes, int n_in,
                              void* d_out, int out_size, void* d_ws, size_t ws_size,
                              hipStream_t stream) {
    (void)n_in; (void)out_size; (void)ws_size;
    const float* X = (const float*)d_in[0];   // [B,64]
    const float* w = (const float*)d_in[1];   // [256,64]
    const float* b = (const float*)d_in[2];   // [256]
    const int B = in_sizes[0] / 64;           // 16384

    float* ws  = (float*)d_ws;
    float* P   = ws;           // 256*64 = 16384 floats
    float* p2s = ws + 16384;   // 256
    float* ssp = ws + 16640;   // 256
    float* an  = ws + 16896;   // 1

    geo_prep<<<1, 256, 0, stream>>>(w, b, P, p2s, ssp, an);

    // 4 waves/block, 32 rows/wave -> 128 rows/block
    geo_main<<<B / 128, 128, 0, stream>>>(X, P, p2s, ssp, an, (float*)d_out);
}